// AttnCalc_33071248179583
// MI455X (gfx1250) — compile-verified
//
#include <hip/hip_runtime.h>
#include <hip/hip_bf16.h>
#include <math.h>
#include <stdint.h>

typedef __attribute__((ext_vector_type(2))) float v2f;
typedef __attribute__((ext_vector_type(8))) float v8f;

#define BSZ 16
#define LSZ 96
#define HSZ 128
#define BW_STRIDE 132           // padded LDS row for weight tile (conflict-free b64)
#define ENC_STRIDE 256          // 64 zeros | 128 data | 64 zeros

// ---------------------------------------------------------------------------
// Async stage of one channel's live weight slice W[l, i, 63, 0..127] into LDS
// via GLOBAL_LOAD_ASYNC_TO_LDS_B128 (ASYNCcnt-tracked, CDNA5 async path).
// 96*128 floats = 3072 float4 transfers = 12 async instructions per wave
// (each instruction moves 16B per lane x 32 lanes).
// ---------------------------------------------------------------------------
__device__ __forceinline__ void stage_weights_async(
    const float* __restrict__ w, int i, float* bwbuf, int t) {
#pragma unroll
  for (int rep = 0; rep < 12; ++rep) {
    const int q  = t + rep * 256;
    const int l  = q >> 5;
    const int k4 = (q & 31) << 2;
    const float* src = w + ((((size_t)l * LSZ + i) * HSZ + 63) * HSZ + k4);
    // Low 32 bits of a flat LDS pointer are the LDS byte offset (aperture rule).
    const unsigned lds = (unsigned)(uintptr_t)(bwbuf + l * BW_STRIDE + k4);
    asm volatile("global_load_async_to_lds_b128 %0, %1, off"
                 :: "v"(lds), "v"(src)
                 : "memory");
  }
}

// ---------------------------------------------------------------------------
// Kernel 1: the big conv as an im2col GEMM on V_WMMA_F32_16X16X4_F32.
// grid = (n in [0,16), g in [0,G)), 256 threads = 8 waves.
// Wave w computes enc_f rows h in [16w,16w+16) x all 96 l, for K-slice
// i in [IC*g, IC*(g+1)), accumulating partial[g][n][h][l].
// Weight tiles are double-buffered in LDS with async loads overlapped
// against the WMMA burst of the previous tile.
// ---------------------------------------------------------------------------
template <int G>
__global__ __launch_bounds__(256) void enc_conv_wmma(
    const float* __restrict__ enc,      // (B, L, H)
    const float* __restrict__ w,        // (L, L, H, H) -- only [:, :, 63, :] is live
    float* __restrict__ partial)        // (G, B, H, L)
{
  constexpr int IC = LSZ / G;                   // channels per K-split
  __shared__ float encp[IC * ENC_STRIDE];       // zero-padded encoder rows
  __shared__ float bw[2][LSZ * BW_STRIDE];      // double-buffered weight stage

  const int n = blockIdx.x;
  const int g = blockIdx.y;
  const int t = (int)threadIdx.x;
  const int lane  = t & 31;
  const int wave  = t >> 5;
  const int m     = lane & 15;     // row-in-tile (M) / col-in-tile (N)
  const int half  = lane >> 4;     // lane half selects K=+2 per ISA layout
  const int hbase = wave << 4;

  // Kick off the first weight stage as early as possible.
  stage_weights_async(w, g * IC, bw[0], t);

  // Zero-pad then fill encoder rows for this channel slice (DS only; does not
  // drain ASYNCcnt).
  for (int idx = t; idx < IC * ENC_STRIDE; idx += 256) encp[idx] = 0.0f;
  __syncthreads();
  for (int idx = t; idx < IC * HSZ; idx += 256) {
    const int i = idx >> 7;
    const int h = idx & (HSZ - 1);
    encp[i * ENC_STRIDE + 64 + h] =
        enc[((size_t)n * LSZ + (size_t)(g * IC + i)) * HSZ + h];
  }

  v8f acc[6] = {};  // 6 N-tiles of 16, 48 accumulator VGPRs

  for (int ic = 0; ic < IC; ++ic) {
    // Overlap: issue next chunk's async loads before computing on this one.
    if (ic + 1 < IC) {
      stage_weights_async(w, g * IC + ic + 1, bw[(ic + 1) & 1], t);
      // 12 newest async ops belong to the next stage; <=12 => current stage
      // is fully resident (async loads complete in order).
      asm volatile("s_wait_asynccnt 12" ::: "memory");
    } else {
      asm volatile("s_wait_asynccnt 0" ::: "memory");
    }
    __syncthreads();  // cross-wave visibility of staged LDS (and encp fill)

    const float* bwc = bw[ic & 1];
    // A(h, k) = encp[i][h + k - 63]; base folds the +64 pad and -63 shift.
    const float* arow = &encp[ic * ENC_STRIDE + 1 + hbase + m + 2 * half];
    for (int kw = 0; kw < HSZ; kw += 4) {
      v2f a;                       // ISA 16x4 f32 A layout: VGPR0=K0/K2, VGPR1=K1/K3
      a.x = arow[kw];
      a.y = arow[kw + 1];
#pragma unroll
      for (int tile = 0; tile < 6; ++tile) {
        const float* brow = &bwc[(tile * 16 + m) * BW_STRIDE + kw + 2 * half];
        v2f b;                     // aligned pair -> ds_load_b64, conflict-free
        b.x = brow[0];
        b.y = brow[1];
        acc[tile] = __builtin_amdgcn_wmma_f32_16x16x4_f32(
            false, a, false, b, (short)0, acc[tile], false, false);
      }
    }
    __syncthreads();  // everyone done reading bw[ic&1] before it is re-staged
  }

  // D layout: VGPR r -> M = r + 8*half, N = lane%16. Store [h][l] (l contiguous).
  float* pb = partial + ((size_t)g * BSZ + n) * (size_t)(HSZ * LSZ);
#pragma unroll
  for (int tile = 0; tile < 6; ++tile) {
    const int l = tile * 16 + m;
#pragma unroll
    for (int r = 0; r < 8; ++r) {
      const int h = hbase + r + 8 * half;
      pb[(size_t)h * LSZ + l] = acc[tile][r];
    }
  }
}

// ---------------------------------------------------------------------------
// Kernel 2: per-batch finalize. dec_f matvec, cvg_f (kw=63 slice) + biases,
// sum K-split partials, tanh, dot with v (LDS float atomics), softmax,
// new_coverage, context.
// ---------------------------------------------------------------------------
template <int G>
__global__ __launch_bounds__(128) void attn_finalize(
    const float* __restrict__ enc,       // (B, L, H)
    const float* __restrict__ hidden,    // (B, H)
    const float* __restrict__ coverage,  // (B, L)
    const float* __restrict__ attn_b,    // (L)
    const float* __restrict__ cvg_w,     // (L, L, 1, H) -- only [..., 63] live
    const float* __restrict__ cvg_b,     // (L)
    const float* __restrict__ dec_w,     // (H, H)
    const float* __restrict__ dec_b,     // (H)
    const float* __restrict__ v,         // (B, H)
    const float* __restrict__ partial,   // (G, B, H, L)
    float* __restrict__ out)             // context|attn|new_coverage
{
  __shared__ float decf[HSZ];
  __shared__ float mixl[LSZ];    // cvg_f[l] + attn_conv_b[l] + cvg_conv_b[l]
  __shared__ float scores[LSZ];
  __shared__ float attn[LSZ];
  __shared__ float invsum;

  const int n = blockIdx.x;
  const int t = (int)threadIdx.x;

  {  // dec_f[h] = hidden[n] . dec_w[h,:] + dec_b[h]
    float s = dec_b[t];
    const float* hr = hidden + (size_t)n * HSZ;
    const float* wr = dec_w + (size_t)t * HSZ;
    for (int k = 0; k < HSZ; ++k) s = fmaf(hr[k], wr[k], s);
    decf[t] = s;
  }
  if (t < LSZ) {  // cvg_f[l] = coverage[n] . cvg_w[l,:,0,63] + biases
    float s = cvg_b[t] + attn_b[t];
    const float* cr = coverage + (size_t)n * LSZ;
    for (int i = 0; i < LSZ; ++i)
      s = fmaf(cr[i], cvg_w[((size_t)t * LSZ + i) * HSZ + 63], s);
    mixl[t] = s;
    scores[t] = 0.0f;
  }
  __syncthreads();

  const float* pn = partial + (size_t)n * HSZ * LSZ;
  const float* vr = v + (size_t)n * HSZ;
  for (int idx = t; idx < HSZ * LSZ; idx += 128) {
    const int h = idx / LSZ;
    const int l = idx - h * LSZ;
    float s = decf[h] + mixl[l];
#pragma unroll
    for (int gg = 0; gg < G; ++gg)
      s += pn[(size_t)gg * BSZ * HSZ * LSZ + (size_t)h * LSZ + l];
    atomicAdd(&scores[l], tanhf(s) * vr[h]);  // ds_add_f32
  }
  __syncthreads();

  if (t == 0) {  // softmax over 96 (trivial)
    float mx = -3.402823466e38f;
    for (int l = 0; l < LSZ; ++l) mx = fmaxf(mx, scores[l]);
    float sum = 0.0f;
    for (int l = 0; l < LSZ; ++l) {
      const float e = __expf(scores[l] - mx);
      attn[l] = e;
      sum += e;
    }
    invsum = 1.0f / sum;
  }
  __syncthreads();
  if (t < LSZ) {
    const float a = attn[t] * invsum;
    attn[t] = a;
    out[BSZ * HSZ + (size_t)n * LSZ + t] = a;                              // attn
    out[BSZ * HSZ + BSZ * LSZ + (size_t)n * LSZ + t] =
        coverage[(size_t)n * LSZ + t] + a;                                 // new_coverage
  }
  __syncthreads();
  {  // context[h] = sum_l attn[l] * enc[n,l,h]
    float c = 0.0f;
    const float* er = enc + (size_t)n * LSZ * HSZ + t;
    for (int l = 0; l < LSZ; ++l) c = fmaf(attn[l], er[(size_t)l * HSZ], c);
    out[(size_t)n * HSZ + t] = c;
  }
}

extern "C" void kernel_launch(void* const* d_in, const int* in_sizes, int n_in,
                              void* d_out, int out_size, void* d_ws, size_t ws_size,
                              hipStream_t stream) {
  (void)in_sizes; (void)n_in; (void)out_size;
  const float* hidden   = (const float*)d_in[0];
  const float* enc      = (const float*)d_in[1];
  const float* coverage = (const float*)d_in[2];
  const float* attn_w   = (const float*)d_in[3];
  const float* attn_b   = (const float*)d_in[4];
  const float* cvg_w    = (const float*)d_in[5];
  const float* cvg_b    = (const float*)d_in[6];
  const float* dec_w    = (const float*)d_in[7];
  const float* dec_b    = (const float*)d_in[8];
  const float* v        = (const float*)d_in[9];
  float* out     = (float*)d_out;
  float* partial = (float*)d_ws;

  const size_t per_g = (size_t)BSZ * HSZ * LSZ * sizeof(float);  // 786 KB
  if (ws_size >= 16 * per_g) {            // 12.6 MB -> 256 workgroups
    hipLaunchKernelGGL((enc_conv_wmma<16>), dim3(BSZ, 16), dim3(256), 0, stream,
                       enc, attn_w, partial);
    hipLaunchKernelGGL((attn_finalize<16>), dim3(BSZ), dim3(128), 0, stream,
                       enc, hidden, coverage, attn_b, cvg_w, cvg_b,
                       dec_w, dec_b, v, partial, out);
  } else if (ws_size >= 8 * per_g) {      // 6.3 MB -> 128 workgroups
    hipLaunchKernelGGL((enc_conv_wmma<8>), dim3(BSZ, 8), dim3(256), 0, stream,
                       enc, attn_w, partial);
    hipLaunchKernelGGL((attn_finalize<8>), dim3(BSZ), dim3(128), 0, stream,
                       enc, hidden, coverage, attn_b, cvg_w, cvg_b,
                       dec_w, dec_b, v, partial, out);
  } else {                                // 3.1 MB fallback
    hipLaunchKernelGGL((enc_conv_wmma<4>), dim3(BSZ, 4), dim3(256), 0, stream,
                       enc, attn_w, partial);
    hipLaunchKernelGGL((attn_finalize<4>), dim3(BSZ), dim3(128), 0, stream,
                       enc, hidden, coverage, attn_b, cvg_w, cvg_b,
                       dec_w, dec_b, v, partial, out);
  }
}